// DeepDeltaResidualExpanded_57552561766784
// MI455X (gfx1250) — compile-verified
//
#include <hip/hip_runtime.h>

typedef float v2f __attribute__((ext_vector_type(2)));
typedef float v8f __attribute__((ext_vector_type(8)));

#define DD 1024
#define DV 4
#define TOK_PER_BLK 16
#define NTHREADS 256
#define NWAVES 8
#define DSLICE (DD / NWAVES)   // 128
#define NSTEPS (DSLICE / 4)    // 32 WMMA K-steps per wave per chain
#define K_SCALE 0.03125f       // 1/sqrt(1024)

__launch_bounds__(NTHREADS)
__global__ void deep_delta_kernel(
    const float* __restrict__ x,
    const float* __restrict__ k_in,
    const float* __restrict__ v_in,
    const float* __restrict__ context,
    const float* __restrict__ beta_w,
    const float* __restrict__ beta_b,
    const float* __restrict__ v_w,
    const float* __restrict__ v_b,
    float* __restrict__ out)
{
    // s_vw holds v_w rows 0..3 plus a zeroed row 4; lanes with col >= 4 read the
    // zero row, so B construction in the WMMA loop is a single ds_load_b64 per
    // chain with no exec masking.
    __shared__ __align__(16) float s_bw[DD];              // beta_w
    __shared__ __align__(16) float s_vw[(DV + 1) * DD];   // v_w rows + zero row
    __shared__ float s_part[NTHREADS];                    // ssq partials
    __shared__ float s_inv[TOK_PER_BLK];
    __shared__ float s_gb[NWAVES * TOK_PER_BLK];          // per-wave beta-logit partials
    __shared__ float s_gv[NWAVES * TOK_PER_BLK * DV];     // per-wave v-logit partials
    __shared__ __align__(16) float s_projp[NTHREADS * DV];
    __shared__ float s_beta[TOK_PER_BLK];
    __shared__ float s_vg[TOK_PER_BLK * DV];
    __shared__ __align__(16) float s_delta[TOK_PER_BLK * DV];

    const int tid  = threadIdx.x;
    const int wave = tid >> 5;
    const int lane = tid & 31;
    const size_t token0 = (size_t)blockIdx.x * TOK_PER_BLK;

    // ---- stage gate weights into LDS (coalesced b128), zero the pad row ----
    ((float4*)s_bw)[tid] = ((const float4*)beta_w)[tid];          // 1024 floats
#pragma unroll
    for (int r = 0; r < 4; ++r)                                   // 4096 floats
        ((float4*)s_vw)[tid + NTHREADS * r] = ((const float4*)v_w)[tid + NTHREADS * r];
    ((float4*)(s_vw + DV * DD))[tid] = make_float4(0.f, 0.f, 0.f, 0.f); // zero row

    // ---- per-token sum of squares for RMS norm ----
    const int tok = tid >> 4;        // token within tile (16 threads/token)
    const int sub = tid & 15;
    const float* kp = k_in + (token0 + tok) * DD;
    float ss = 0.f;
#pragma unroll
    for (int j = 0; j < 16; ++j) {
        float4 kv = ((const float4*)kp)[sub + 16 * j];
        ss += kv.x * kv.x + kv.y * kv.y + kv.z * kv.z + kv.w * kv.w;
    }
    s_part[tid] = ss;
    __syncthreads();
    if (tid < TOK_PER_BLK) {
        float s = 0.f;
#pragma unroll
        for (int j = 0; j < 16; ++j) s += s_part[tid * 16 + j];
        const float eps_rms = (1e-5f * 1e-5f) / (float)DD;
        s_inv[tid] = rsqrtf(s * (1.0f / (float)DD) + eps_rms);
    }
    __syncthreads();   // s_inv, s_bw, s_vw (incl. zero row) visible to all waves

    // ---- fp32 WMMA chains: beta logits (context . beta_w), v logits (v_in . v_w^T)
    // A (16x4 f32): lane l holds row m = l&15, k = 2*(l>>4) + vgpr
    // B (4x16 f32): lane l holds col n = l&15, k = 2*(l>>4) + vgpr
    // C (16x16 f32): vgpr i on lane l holds M = i + 8*(l>>4), N = l&15
    const int rowm  = lane & 15;
    const int half  = lane >> 4;
    const int kbase = half * 2;
    const int d_lo  = wave * DSLICE;

    const float* aB = context + (token0 + rowm) * DD + kbase + d_lo;
    const float* aV = v_in    + (token0 + rowm) * DD + kbase + d_lo;
    // branch-free per-lane B pointers (computed once)
    const float* bB = ((rowm == 0) ? s_bw : (s_vw + DV * DD)) + kbase + d_lo;
    const float* bV = s_vw + ((rowm < DV) ? rowm : DV) * DD + kbase + d_lo;

    v8f cb = {};
    v8f cv = {};
#pragma unroll 4
    for (int s = 0; s < NSTEPS; ++s) {
        v2f a_b = *(const v2f*)(aB + s * 4);
        v2f a_v = *(const v2f*)(aV + s * 4);
        v2f b_b = *(const v2f*)(bB + s * 4);
        v2f b_v = *(const v2f*)(bV + s * 4);
        cb = __builtin_amdgcn_wmma_f32_16x16x4_f32(false, a_b, false, b_b,
                                                   (short)0, cb, false, false);
        cv = __builtin_amdgcn_wmma_f32_16x16x4_f32(false, a_v, false, b_v,
                                                   (short)0, cv, false, false);
    }

    // ---- deterministic cross-wave combine of C fragments via LDS ----
    // Hoisted guards: one branch region per condition, contiguous ds_stores inside.
    if (rowm == 0) {
        float* g = s_gb + wave * TOK_PER_BLK + 8 * half;
#pragma unroll
        for (int i = 0; i < 8; ++i) g[i] = cb[i];
    }
    if (rowm < DV) {
        float* g = s_gv + wave * (TOK_PER_BLK * DV) + (8 * half) * DV + rowm;
#pragma unroll
        for (int i = 0; i < 8; ++i) g[i * DV] = cv[i];
    }
    __syncthreads();

    if (tid < TOK_PER_BLK) {
        float acc = beta_b[0];
#pragma unroll
        for (int w = 0; w < NWAVES; ++w) acc += s_gb[w * TOK_PER_BLK + tid];
        s_beta[tid] = 2.0f / (1.0f + __expf(-acc));
    }
    if (tid < TOK_PER_BLK * DV) {
        const int n = tid & 3;
        float acc = v_b[n];
#pragma unroll
        for (int w = 0; w < NWAVES; ++w) acc += s_gv[w * (TOK_PER_BLK * DV) + tid];
        s_vg[tid] = 4.0f / (1.0f + __expf(-acc));   // sigmoid * V_SIG_SCALE
    }
    __syncthreads();

    // ---- pass 1 over x: proj[t][v] = sum_d k_in*x  (b128 streaming loads) ----
    const float* xRow = x + (token0 + tok) * (size_t)(DD * DV);
    float4 acc = make_float4(0.f, 0.f, 0.f, 0.f);
#pragma unroll 8
    for (int j = 0; j < 64; ++j) {
        const int d = sub + 16 * j;
        float4 xv = ((const float4*)xRow)[d];
        float  kk = kp[d];
        acc.x += kk * xv.x; acc.y += kk * xv.y;
        acc.z += kk * xv.z; acc.w += kk * xv.w;
    }
    ((float4*)s_projp)[tid] = acc;
    __syncthreads();

    if (tid < TOK_PER_BLK * DV) {
        const int m = tid >> 2, n = tid & 3;
        float p = 0.f;
#pragma unroll
        for (int s = 0; s < 16; ++s) p += s_projp[(m * 16 + s) * DV + n];
        const float pf = p * s_inv[m] * K_SCALE;
        s_delta[tid] = s_beta[m] * (s_vg[tid] - pf) * K_SCALE;
    }
    __syncthreads();

    // ---- pass 2 over x (L2-hot re-read): out = x + k_rms * delta_row ----
    const float4 dl  = ((const float4*)s_delta)[tok];
    const float  inv = s_inv[tok];
    float* oRow = out + (token0 + tok) * (size_t)(DD * DV);
#pragma unroll 8
    for (int j = 0; j < 64; ++j) {
        const int d = sub + 16 * j;
        float4 xv = ((const float4*)xRow)[d];
        const float r = kp[d] * inv;
        float4 ov;
        ov.x = xv.x + r * dl.x; ov.y = xv.y + r * dl.y;
        ov.z = xv.z + r * dl.z; ov.w = xv.w + r * dl.w;
        ((float4*)oRow)[d] = ov;
    }
}

extern "C" void kernel_launch(void* const* d_in, const int* in_sizes, int n_in,
                              void* d_out, int out_size, void* d_ws, size_t ws_size,
                              hipStream_t stream) {
    const float* x       = (const float*)d_in[0];
    const float* k_in    = (const float*)d_in[1];
    const float* v_in    = (const float*)d_in[2];
    const float* context = (const float*)d_in[3];
    const float* beta_w  = (const float*)d_in[4];
    const float* beta_b  = (const float*)d_in[5];
    const float* v_w     = (const float*)d_in[6];
    const float* v_b     = (const float*)d_in[7];

    const int ntok  = in_sizes[1] / DD;          // B*T = 16384
    const int nblks = ntok / TOK_PER_BLK;        // 1024
    deep_delta_kernel<<<nblks, NTHREADS, 0, stream>>>(
        x, k_in, v_in, context, beta_w, beta_b, v_w, v_b, (float*)d_out);
}